// UltimateSmoother_15625091023326
// MI455X (gfx1250) — compile-verified
//
#include <hip/hip_runtime.h>

typedef __attribute__((ext_vector_type(2))) float v2f;
typedef __attribute__((ext_vector_type(8))) float v8f;

// UltimateSmoother: 2nd-order IIR along T for X[B=32][T=8192][C=256] fp32.
// y[t] = k1*x[t-1] + k2*x[t-2] + k3*x[t-3] + c2*y[t-1] + c3*y[t-2], y[0..2]=x[0..2].
//
// Tile formulation: 16 outputs y[t0..t0+15] are an exact linear map of
// v = [ x[t0-3..t0+14] (18 rows), y[t0-1], y[t0-2] ]  (20 rows total)
// via a 16x20 matrix G built from the same recurrence. K padded to 20 = 5*4:
// five chained V_WMMA_F32_16X16X4_F32 per tile (16 timesteps x 16 channels).
// Hot loop = 511 full tiles (unconditional, unrolled x2 for load/WMMA overlap);
// the partial tail tile (13 outputs) is peeled with guarded stores.

__global__ __launch_bounds__(32)
void ultimate_smoother_wmma(const float* __restrict__ X,
                            const float* __restrict__ periodp,
                            float* __restrict__ Y)
{
    constexpr int    T    = 8192;
    constexpr int    C    = 256;
    constexpr size_t TILE = (size_t)16 * C;   // elements per 16-timestep tile

    const int lane = threadIdx.x;      // 0..31 (wave32)
    const int m    = lane & 15;        // A row / B,C column
    const int hi   = lane >> 4;        // half-wave selector
    const int s    = hi * 2;           // K-pair selector per ISA A/B layout

    const int chg = blockIdx.x;        // 0..15 channel group
    const int b   = blockIdx.y;        // 0..31 batch
    const int ch  = chg * 16 + m;

    // --- coefficients from device scalar `period` (matches reference) ---
    const float period = periodp[0];
    const float SQ2    = 1.41421356237309515f;
    const float SQ2PI  = 4.44288293815836625f;           // sqrt(2)*pi
    const float a1 = __expf(-SQ2PI / period);
    const float c3 = -a1 * a1;
    const float c2 = -2.0f * a1 * __cosf(SQ2 * 180.0f / period);
    const float c1 = (1.0f + c2 - c3) * 0.25f;
    const float k1 = 1.0f - c1;
    const float k2 = 2.0f * c1 - c2;
    const float k3 = -(c1 + c3);
    const float r1 = c2;
    const float r2 = c3;

    // --- build G (16 x 20) in LDS: column j owned by lane j ---
    // g_i = k1*e_{i+2} + k2*e_{i+1} + k3*e_i + r1*g_{i-1} + r2*g_{i-2}
    // with g_{-1} = e18 (y[t0-1]), g_{-2} = e19 (y[t0-2]).
    __shared__ float Gs[16][20];
    if (lane < 20) {
        const int j = lane;
        float gm1 = (j == 18) ? 1.0f : 0.0f;
        float gm2 = (j == 19) ? 1.0f : 0.0f;
#pragma unroll
        for (int i = 0; i < 16; ++i) {
            float gi = r1 * gm1 + r2 * gm2;
            gi += (j == i + 2) ? k1 : 0.0f;
            gi += (j == i + 1) ? k2 : 0.0f;
            gi += (j == i    ) ? k3 : 0.0f;
            Gs[i][j] = gi;
            gm2 = gm1;
            gm1 = gi;
        }
    }
    __syncthreads();

    // --- A operands in registers, per ISA 32-bit 16x4 A layout ---
    v2f A[5];
#pragma unroll
    for (int k = 0; k < 5; ++k) {
        A[k].x = Gs[m][4 * k + s];
        A[k].y = Gs[m][4 * k + s + 1];
    }

    const float* __restrict__ Xp = X + (size_t)b * T * C + ch;
    float*       __restrict__ Yp = Y + (size_t)b * T * C + ch;

    // --- passthrough y[0..2] = x[0..2] ---
    if (lane < 16) {
        Yp[0 * C] = Xp[0 * C];
        Yp[1 * C] = Xp[1 * C];
        Yp[2 * C] = Xp[2 * C];
    }

    // --- initial carried state: y[2] = x[2], y[1] = x[1] ---
    float yp1 = Xp[2 * C];   // y[t0-1]
    float yp2 = Xp[1 * C];   // y[t0-2]

    // rolling pointers: xt -> x[t0-3][ch], yt -> y[t0 + hi*8][ch]
    const float* __restrict__ xt = Xp;                         // t0 = 3
    float*       __restrict__ yt = Yp + (3 + hi * 8) * C;

    // ---------------- hot loop: 511 full tiles, t0 = 3 .. 8163 ----------------
#pragma unroll 2
    for (int iter = 0; iter < 511; ++iter) {
        // speculative stream-prefetch two tiles ahead (global_prefetch_b8);
        // OOB overshoot on the last tiles is silently dropped (TH=0 speculative)
        __builtin_prefetch(xt + 32 * (size_t)C, 0, 1);

        v8f acc = {};

        // chunks 0..3: pure-input K rows 0..15 -> x[t0-3 .. t0+12]
#pragma unroll
        for (int k = 0; k < 4; ++k) {
            v2f Bk;
            Bk.x = xt[(size_t)(4 * k + s) * C];
            Bk.y = xt[(size_t)(4 * k + s + 1) * C];
            acc = __builtin_amdgcn_wmma_f32_16x16x4_f32(
                false, A[k], false, Bk, (short)0, acc, false, false);
        }

        // chunk 4: K rows {16,17} = x[t0+13], x[t0+14] (lanes 0-15);
        //          K rows {18,19} = y[t0-1], y[t0-2]   (lanes 16-31)
        const float xa = xt[16 * (size_t)C];
        const float xb = xt[17 * (size_t)C];
        v2f B4;
        B4.x = hi ? yp1 : xa;
        B4.y = hi ? yp2 : xb;
        acc = __builtin_amdgcn_wmma_f32_16x16x4_f32(
            false, A[4], false, B4, (short)0, acc, false, false);

        // store D: element r -> timestep t0 + r + hi*8, channel m (all in-bounds)
#pragma unroll
        for (int r = 0; r < 8; ++r) {
            yt[(size_t)r * C] = acc[r];
        }

        // carry state: y[t0+15] = D row15 (lanes 16-31, VGPR7),
        //              y[t0+14] = D row14 (lanes 16-31, VGPR6)
        yp1 = __shfl(acc[7], 16 + m, 32);
        yp2 = __shfl(acc[6], 16 + m, 32);

        xt += TILE;
        yt += TILE;
    }

    // ---------------- peeled tail tile: t0 = 8179, outputs 8179..8191 ----------------
    {
        v8f acc = {};
#pragma unroll
        for (int k = 0; k < 4; ++k) {           // rows 0..15 -> t = 8176..8191, in-bounds
            v2f Bk;
            Bk.x = xt[(size_t)(4 * k + s) * C];
            Bk.y = xt[(size_t)(4 * k + s + 1) * C];
            acc = __builtin_amdgcn_wmma_f32_16x16x4_f32(
                false, A[k], false, Bk, (short)0, acc, false, false);
        }
        // rows 16,17 would be x[8192],x[8193]: clamp to x[8191]; they only feed
        // outputs y[8193],y[8194] which are never stored.
        const float xc = xt[15 * (size_t)C];
        v2f B4;
        B4.x = hi ? yp1 : xc;
        B4.y = hi ? yp2 : xc;
        acc = __builtin_amdgcn_wmma_f32_16x16x4_f32(
            false, A[4], false, B4, (short)0, acc, false, false);

        constexpr int t0 = 8179;
#pragma unroll
        for (int r = 0; r < 8; ++r) {
            const int t = t0 + r + hi * 8;
            if (t < T) {
                yt[(size_t)r * C] = acc[r];
            }
        }
    }
}

extern "C" void kernel_launch(void* const* d_in, const int* in_sizes, int n_in,
                              void* d_out, int out_size, void* d_ws, size_t ws_size,
                              hipStream_t stream) {
    const float* X      = (const float*)d_in[0];
    const float* period = (const float*)d_in[1];
    float*       Y      = (float*)d_out;
    (void)in_sizes; (void)n_in; (void)out_size; (void)d_ws; (void)ws_size;

    dim3 grid(16, 32);   // 16 channel groups x 32 batches = 512 single-wave blocks
    dim3 block(32);      // one wave32 per block
    hipLaunchKernelGGL(ultimate_smoother_wmma, grid, block, 0, stream, X, period, Y);
}